// FusionFeature_24988119728842
// MI455X (gfx1250) — compile-verified
//
#include <hip/hip_runtime.h>
#include <math.h>

#define NCELL 4096
#define GDIM  2048
#define CDIM  2048
#define MDIM  1024
#define NDRUG 2048
#define DDIM  1024
#define SHDIM 128
#define ATTD  32
#define KTOP  10
#define EPSV  1e-8f

typedef float v2f __attribute__((ext_vector_type(2)));
typedef float v8f __attribute__((ext_vector_type(8)));

// ---------------------------------------------------------------- utilities
__global__ void zero_floats(float* p, int n) {
    int i = blockIdx.x * blockDim.x + threadIdx.x;
    if (i < n) p[i] = 0.f;
}

// Phase 1 of column z-norm: partial column sums / sums-of-squares.
// Block = 64 columns x 4 row-threads, grid.y splits the rows. Coalesced.
__global__ void colstat_partial(const float* __restrict__ A, int M, int K,
                                float* __restrict__ csum, float* __restrict__ csq) {
    __shared__ float s0[256];
    __shared__ float s1[256];
    int cl = threadIdx.x & 63;
    int rs = threadIdx.x >> 6;                 // 0..3
    int c  = blockIdx.x * 64 + cl;
    int rows_per = M / gridDim.y;
    int r0 = blockIdx.y * rows_per;
    float s = 0.f, ss = 0.f;
    for (int r = r0 + rs; r < r0 + rows_per; r += 4) {
        float x = A[(size_t)r * K + c];
        s += x; ss += x * x;
    }
    s0[threadIdx.x] = s; s1[threadIdx.x] = ss;
    __syncthreads();
    if (rs == 0) {
        float t0 = s0[cl] + s0[cl + 64] + s0[cl + 128] + s0[cl + 192];
        float t1 = s1[cl] + s1[cl + 64] + s1[cl + 128] + s1[cl + 192];
        atomicAdd(&csum[c], t0);
        atomicAdd(&csq[c],  t1);
    }
}

// Phase 2: out = (x - mean_c) / (std_c + eps), unbiased std (ddof=1).
__global__ void znorm_apply(const float* __restrict__ A, float* __restrict__ out,
                            const float* __restrict__ csum,
                            const float* __restrict__ csq, int M, int K) {
    size_t i = (size_t)blockIdx.x * blockDim.x + threadIdx.x;
    if (i >= (size_t)M * K) return;
    int c = (int)(i % K);
    float mean = csum[c] / (float)M;
    float var  = (csq[c] - (float)M * mean * mean) / (float)(M - 1);
    var = fmaxf(var, 0.f);
    float x = A[i];
    out[i] = (x - mean) / (sqrtf(var) + EPSV);
}

// Per-row reduction: sum (square=0) or sum of squares (square=1).
__global__ void row_reduce(const float* __restrict__ A, int K,
                           float* __restrict__ out, int square) {
    __shared__ float red[256];
    int row = blockIdx.x;
    const float* a = A + (size_t)row * K;
    float s = 0.f;
    for (int k = threadIdx.x; k < K; k += blockDim.x) {
        float x = a[k];
        s += square ? x * x : x;
    }
    red[threadIdx.x] = s;
    __syncthreads();
    for (int st = 128; st > 0; st >>= 1) {
        if ((int)threadIdx.x < st) red[threadIdx.x] += red[threadIdx.x + st];
        __syncthreads();
    }
    if (threadIdx.x == 0) out[row] = red[0];
}

// ------------------------------- symmetric Gram matrix via f32 WMMA, 4x2 reg
// C = transform(A @ A.T). Upper-triangular 64x64 macro tiles; each macro tile
// is split between two waves (64x32 halves). Off-diagonal macro tiles are
// mirrored. 8 independent WMMA chains per wave; K and MODE are compile-time
// so fragment addresses fold into immediate offsets (no spills).
// MODE 0: RBF (aux=row|x|^2)  MODE 1: (g+1)^3  MODE 2: jaccard (aux=row-sum)
template <int K, int MODE>
__global__ __launch_bounds__(256, 1)
void gram_wmma_sym(const float* __restrict__ A, int N,
                   float* __restrict__ out,
                   const float* __restrict__ aux,
                   float* __restrict__ sumsq_p) {
    int wave = threadIdx.x >> 5;
    int lane = threadIdx.x & 31;
    int Mt = N >> 6;                           // 64-wide macro tiles per side
    int T  = Mt * (Mt + 1) / 2;
    int w  = blockIdx.x * (blockDim.x >> 5) + wave;
    int t  = w >> 1;                           // macro tile index
    int nh = w & 1;                            // which 32-col half
    if (t >= T) return;                        // wave-uniform
    // decode (TM <= TN) from linear triangular index; S(i)=i*Mt-i*(i-1)/2
    float fb = 2.f * (float)Mt + 1.f;
    int TM = (int)((fb - sqrtf(fb * fb - 8.f * (float)t)) * 0.5f);
    if (TM < 0) TM = 0;
    if (TM > Mt - 1) TM = Mt - 1;
    while (TM > 0 && (TM * Mt - TM * (TM - 1) / 2) > t) --TM;
    while (((TM + 1) * Mt - (TM + 1) * TM / 2) <= t) ++TM;
    int TN = TM + (t - (TM * Mt - TM * (TM - 1) / 2));

    int half = lane >> 4, l16 = lane & 15;
    const float* aB = A + (size_t)(TM * 64 + l16) * K;
    const float* bB = A + (size_t)(TN * 64 + nh * 32 + l16) * K;

    v8f acc[8];
#pragma unroll
    for (int i = 0; i < 8; ++i)
        acc[i] = (v8f){0.f, 0.f, 0.f, 0.f, 0.f, 0.f, 0.f, 0.f};

    for (int k = 0; k < K; k += 4) {
        int kk = k + half * 2;
        v2f af[4], bf[2];
#pragma unroll
        for (int i = 0; i < 4; ++i) {
            af[i].x = aB[i * 16 * K + kk];
            af[i].y = aB[i * 16 * K + kk + 1];
        }
#pragma unroll
        for (int j = 0; j < 2; ++j) {
            bf[j].x = bB[j * 16 * K + kk];
            bf[j].y = bB[j * 16 * K + kk + 1];
        }
#pragma unroll
        for (int im = 0; im < 4; ++im)
#pragma unroll
            for (int in = 0; in < 2; ++in)
                acc[im * 2 + in] = __builtin_amdgcn_wmma_f32_16x16x4_f32(
                    false, af[im], false, bf[in], (short)0,
                    acc[im * 2 + in], false, false);
    }

    int mirror = (TM != TN);
    float lsum = 0.f;
#pragma unroll
    for (int im = 0; im < 4; ++im) {
#pragma unroll
        for (int in = 0; in < 2; ++in) {
            int colb = TN * 64 + nh * 32 + in * 16 + l16;
            int rowb = TM * 64 + im * 16 + half * 8;
            float auxc = (MODE != 1) ? aux[colb] : 0.f;
#pragma unroll
            for (int r = 0; r < 8; ++r) {
                int row = rowb + r;
                float g = acc[im * 2 + in][r];
                float v;
                if (MODE == 0) {
                    float d = fmaxf(aux[row] + auxc - 2.f * g, 0.f);
                    v = expf(-d * 0.125f);     // exp(-d^2/(2*sigma^2)), sigma=2
                } else if (MODE == 1) {
                    float tv = g + 1.f;
                    v = tv * tv * tv;          // polynomial kernel, degree 3
                } else {
                    v = g / (aux[row] + auxc - g + EPSV);  // jaccard
                }
                out[(size_t)row * N + colb] = v;
                if (mirror) out[(size_t)colb * N + row] = v;
                lsum += v * v;
            }
        }
    }
    if (sumsq_p) {
        if (mirror) lsum *= 2.f;
        __shared__ float red[256];
        red[threadIdx.x] = lsum;
        __syncthreads();
        for (int st = 128; st > 0; st >>= 1) {
            if ((int)threadIdx.x < st) red[threadIdx.x] += red[threadIdx.x + st];
            __syncthreads();
        }
        if (threadIdx.x == 0) atomicAdd(sumsq_p, red[0]);
    }
}

// ---------------- out = scale*(sum_t A_t@W_t + sum biases); 4x1 M-blocking,
// B fragment reused across the 4 row tiles.
__global__ __launch_bounds__(256, 1)
void linear_wmma3(const float* A0, const float* W0, int K0,
                  const float* A1, const float* W1, int K1,
                  const float* A2, const float* W2, int K2,
                  const float* b0, const float* b1, const float* b2,
                  float scale, float* __restrict__ out, int M, int N) {
    int wave = threadIdx.x >> 5;
    int lane = threadIdx.x & 31;
    int Nt = N >> 4;
    int t = blockIdx.x * (blockDim.x >> 5) + wave;
    if (t >= (M >> 6) * Nt) return;
    int tm = t / Nt, tn = t % Nt;
    int half = lane >> 4, l16 = lane & 15;
    v8f acc[4];
#pragma unroll
    for (int i = 0; i < 4; ++i)
        acc[i] = (v8f){0.f, 0.f, 0.f, 0.f, 0.f, 0.f, 0.f, 0.f};
    const float* As[3] = {A0, A1, A2};
    const float* Wt[3] = {W0, W1, W2};
    int Ks[3] = {K0, K1, K2};
    for (int q = 0; q < 3; ++q) {
        if (!As[q]) continue;
        int K = Ks[q];
        const float* W = Wt[q];
        const float* ar0 = As[q] + (size_t)(tm * 64 + l16) * K;
        for (int k = 0; k < K; k += 4) {
            int kk = k + half * 2;
            v2f bf;
            bf.x = W[(size_t)kk * N + tn * 16 + l16];
            bf.y = W[(size_t)(kk + 1) * N + tn * 16 + l16];
#pragma unroll
            for (int im = 0; im < 4; ++im) {
                v2f af;
                af.x = ar0[(size_t)im * 16 * K + kk];
                af.y = ar0[(size_t)im * 16 * K + kk + 1];
                acc[im] = __builtin_amdgcn_wmma_f32_16x16x4_f32(
                    false, af, false, bf, (short)0, acc[im], false, false);
            }
        }
    }
    int col = tn * 16 + l16;
    float bc = 0.f;
    if (b0) bc += b0[col];
    if (b1) bc += b1[col];
    if (b2) bc += b2[col];
#pragma unroll
    for (int im = 0; im < 4; ++im) {
        int rowbase = tm * 64 + im * 16 + half * 8;
#pragma unroll
        for (int r = 0; r < 8; ++r)
            out[(size_t)(rowbase + r) * N + col] = (acc[im][r] + bc) * scale;
    }
}

// --------------------------- dyn attention: sum of per-row softmax(MLP(x))[3]
__global__ void dyn_mean(const float* __restrict__ X, const float* W1,
                         const float* b1, const float* W2, const float* b2,
                         float* __restrict__ dynsum, int M) {
    int row = blockIdx.x * blockDim.x + threadIdx.x;
    if (row >= M) return;
    const float* x = X + (size_t)row * SHDIM;
    float h[ATTD];
#pragma unroll
    for (int j = 0; j < ATTD; ++j) h[j] = b1[j];
    for (int k = 0; k < SHDIM; ++k) {
        float xv = x[k];
        const float* wr = W1 + k * ATTD;
#pragma unroll
        for (int j = 0; j < ATTD; ++j) h[j] += xv * wr[j];
    }
    float z0 = b2[0], z1 = b2[1], z2 = b2[2];
#pragma unroll
    for (int j = 0; j < ATTD; ++j) {
        float hv = fmaxf(h[j], 0.f);
        z0 += hv * W2[j * 3 + 0];
        z1 += hv * W2[j * 3 + 1];
        z2 += hv * W2[j * 3 + 2];
    }
    float mx = fmaxf(z0, fmaxf(z1, z2));
    float e0 = expf(z0 - mx), e1 = expf(z1 - mx), e2 = expf(z2 - mx);
    float inv = 1.f / (e0 + e1 + e2);
    atomicAdd(&dynsum[0], e0 * inv);
    atomicAdd(&dynsum[1], e1 * inv);
    atomicAdd(&dynsum[2], e2 * inv);
}

// ----------------------- neighbor attention scores: sigmoid(MLP(x)) per row
// (sum-normalization skipped: a positive rescale never changes top-k
//  selection, and kept values come from the raw sim matrix)
__global__ void att_score(const float* __restrict__ X, int M,
                          const float* W1, const float* b1,
                          const float* W2, const float* b2,
                          float* __restrict__ scores) {
    int row = blockIdx.x * blockDim.x + threadIdx.x;
    if (row >= M) return;
    const float* x = X + (size_t)row * SHDIM;
    float h[ATTD];
#pragma unroll
    for (int j = 0; j < ATTD; ++j) h[j] = b1[j];
    for (int k = 0; k < SHDIM; ++k) {
        float xv = x[k];
        const float* wr = W1 + k * ATTD;
#pragma unroll
        for (int j = 0; j < ATTD; ++j) h[j] += xv * wr[j];
    }
    float z = b2[0];
#pragma unroll
    for (int j = 0; j < ATTD; ++j) z += fmaxf(h[j], 0.f) * W2[j];
    scores[row] = 1.f / (1.f + expf(-z));
}

// coef[j] = (0.5*softmax(static)[j] + 0.5*mean(dyn)[j]) / (||k_j||_F + eps)
__global__ void finalize_coef(const float* sw, const float* dynsum,
                              const float* sumsq, float* coef, float invM) {
    if (threadIdx.x == 0 && blockIdx.x == 0) {
        float mx = fmaxf(sw[0], fmaxf(sw[1], sw[2]));
        float e0 = expf(sw[0] - mx), e1 = expf(sw[1] - mx), e2 = expf(sw[2] - mx);
        float inv = 1.f / (e0 + e1 + e2);
        float stat[3] = {e0 * inv, e1 * inv, e2 * inv};
        for (int j = 0; j < 3; ++j) {
            float mixed = 0.5f * stat[j] + 0.5f * dynsum[j] * invM;
            coef[j] = mixed / (sqrtf(sumsq[j]) + EPSV);
        }
    }
}

// ----------------- fuse kernels on the fly, per-row top-10, sparse @ Wk + bk
__global__ void topk_feature(const float* __restrict__ s0,
                             const float* __restrict__ s1,
                             const float* __restrict__ s2,
                             const float* __restrict__ coef,
                             const float* __restrict__ scores,
                             const float* __restrict__ Wk,
                             const float* __restrict__ bk,
                             float* __restrict__ out, int N, int row_normalize) {
    __shared__ float fusedS[NCELL];
    __shared__ float wvS[NCELL];
    __shared__ float rv[256];
    __shared__ int   ri[256];
    __shared__ float selv[KTOP];
    __shared__ int   seli[KTOP];
    int row = blockIdx.x;
    int tid = threadIdx.x;
    float c0 = 0.f, c1 = 0.f, c2 = 0.f;
    if (s1) { c0 = coef[0]; c1 = coef[1]; c2 = coef[2]; }
    for (int j = tid; j < N; j += blockDim.x) {
        size_t p = (size_t)row * N + j;
        float f = s1 ? (c0 * s0[p] + c1 * s1[p] + c2 * s2[p]) : s0[p];
        fusedS[j] = f;
        wvS[j] = f * scores[j];
    }
    __syncthreads();
    for (int t = 0; t < KTOP; ++t) {
        float bv = -INFINITY; int bi = N;
        for (int j = tid; j < N; j += blockDim.x) {
            float v = wvS[j];
            if (v > bv || (v == bv && j < bi)) { bv = v; bi = j; }
        }
        rv[tid] = bv; ri[tid] = bi;
        __syncthreads();
        for (int st = 128; st > 0; st >>= 1) {
            if (tid < st) {
                float ov = rv[tid + st]; int oi = ri[tid + st];
                if (ov > rv[tid] || (ov == rv[tid] && oi < ri[tid])) {
                    rv[tid] = ov; ri[tid] = oi;
                }
            }
            __syncthreads();
        }
        if (tid == 0) {
            int idx = ri[0];
            seli[t] = idx;
            selv[t] = fusedS[idx];
            wvS[idx] = -INFINITY;
        }
        __syncthreads();
    }
    if (row_normalize) {
        if (tid == 0) {
            float s = 0.f;
            for (int t = 0; t < KTOP; ++t) s += selv[t];
            float inv = 1.f / (s + EPSV);
            for (int t = 0; t < KTOP; ++t) selv[t] *= inv;
        }
        __syncthreads();
    }
    for (int col = tid; col < SHDIM; col += blockDim.x) {
        float acc = bk[col];
#pragma unroll
        for (int t = 0; t < KTOP; ++t)
            acc += selv[t] * Wk[(size_t)seli[t] * SHDIM + col];
        out[(size_t)row * SHDIM + col] = acc;
    }
}

// --------------------------------------------------------------------------
extern "C" void kernel_launch(void* const* d_in, const int* in_sizes, int n_in,
                              void* d_out, int out_size, void* d_ws, size_t ws_size,
                              hipStream_t stream) {
    (void)in_sizes; (void)n_in; (void)out_size; (void)ws_size;
    const float* gene_raw = (const float*)d_in[0];
    const float* cna_raw  = (const float*)d_in[1];
    const float* mutation = (const float*)d_in[2];
    const float* fdrug    = (const float*)d_in[3];
    const float* Wg  = (const float*)d_in[4];  const float* bg  = (const float*)d_in[5];
    const float* Wc  = (const float*)d_in[6];  const float* bc  = (const float*)d_in[7];
    const float* Wm  = (const float*)d_in[8];  const float* bm  = (const float*)d_in[9];
    const float* Wd  = (const float*)d_in[10]; const float* bd  = (const float*)d_in[11];
    const float* statw = (const float*)d_in[12];
    const float* Wa1 = (const float*)d_in[13]; const float* ba1 = (const float*)d_in[14];
    const float* Wa2 = (const float*)d_in[15]; const float* ba2 = (const float*)d_in[16];
    const float* Wca1 = (const float*)d_in[17]; const float* bca1 = (const float*)d_in[18];
    const float* Wca2 = (const float*)d_in[19]; const float* bca2 = (const float*)d_in[20];
    const float* Wda1 = (const float*)d_in[21]; const float* bda1 = (const float*)d_in[22];
    const float* Wda2 = (const float*)d_in[23]; const float* bda2 = (const float*)d_in[24];
    const float* WkC = (const float*)d_in[25]; const float* bkC = (const float*)d_in[26];
    const float* WkD = (const float*)d_in[27]; const float* bkD = (const float*)d_in[28];
    float* out = (float*)d_out;
    float* ws  = (float*)d_ws;

    // workspace layout (floats); total ~72.1M floats (~289 MB)
    float* geneN  = ws;
    float* cnaN   = geneN  + (size_t)NCELL * GDIM;
    float* k0     = cnaN   + (size_t)NCELL * CDIM;
    float* k1     = k0     + (size_t)NCELL * NCELL;
    float* k2     = k1     + (size_t)NCELL * NCELL;
    float* dsim   = k2     + (size_t)NCELL * NCELL;
    float* shrd   = dsim   + (size_t)NDRUG * NDRUG;
    float* dembed = shrd   + (size_t)NCELL * SHDIM;
    float* geneSq = dembed + (size_t)NDRUG * SHDIM;
    float* mutSum = geneSq + NCELL;
    float* drugSum= mutSum + NCELL;
    float* cellSc = drugSum+ NDRUG;
    float* drugSc = cellSc + NCELL;
    float* colSumG= drugSc + NDRUG;     // 2048
    float* colSqG = colSumG + GDIM;     // 2048
    float* colSumC= colSqG  + GDIM;     // 2048
    float* colSqC = colSumC + CDIM;     // 2048
    float* scal   = colSqC  + CDIM;     // [0..2]=sumsq, [3..5]=dynsum, [6..8]=coef
    float* sumsq  = scal;
    float* dynsum = scal + 3;
    float* coef   = scal + 6;

    // zero column stats + scalar accumulators (graph-safe, every call)
    zero_floats<<<(4 * GDIM + 9 + 255) / 256, 256, 0, stream>>>(colSumG, 4 * GDIM + 9);

    // z-normalization (two-phase, fully parallel, coalesced)
    colstat_partial<<<dim3(GDIM / 64, 32), 256, 0, stream>>>(gene_raw, NCELL, GDIM, colSumG, colSqG);
    colstat_partial<<<dim3(CDIM / 64, 32), 256, 0, stream>>>(cna_raw,  NCELL, CDIM, colSumC, colSqC);
    znorm_apply<<<(NCELL * GDIM) / 256, 256, 0, stream>>>(gene_raw, geneN, colSumG, colSqG, NCELL, GDIM);
    znorm_apply<<<(NCELL * CDIM) / 256, 256, 0, stream>>>(cna_raw,  cnaN,  colSumC, colSqC, NCELL, CDIM);

    row_reduce<<<NCELL, 256, 0, stream>>>(geneN,    GDIM, geneSq,  1);
    row_reduce<<<NCELL, 256, 0, stream>>>(mutation, MDIM, mutSum,  0);
    row_reduce<<<NDRUG, 256, 0, stream>>>(fdrug,    DDIM, drugSum, 0);

    {   // cell kernels: Mt=64 macro tiles -> T=2080, 2 waves per tile -> 520 WGs
        int Mt = NCELL / 64;
        int blocks = (2 * (Mt * (Mt + 1) / 2) + 7) / 8;   // 520
        gram_wmma_sym<GDIM, 0><<<blocks, 256, 0, stream>>>(geneN,    NCELL, k0, geneSq, &sumsq[0]);
        gram_wmma_sym<CDIM, 1><<<blocks, 256, 0, stream>>>(cnaN,     NCELL, k1, geneSq, &sumsq[1]);
        gram_wmma_sym<MDIM, 2><<<blocks, 256, 0, stream>>>(mutation, NCELL, k2, mutSum, &sumsq[2]);
    }
    {   // drug jaccard: Mt=32 -> T=528, 2 waves per tile -> 132 WGs
        int Mt = NDRUG / 64;
        int blocks = (2 * (Mt * (Mt + 1) / 2) + 7) / 8;   // 132
        gram_wmma_sym<DDIM, 2><<<blocks, 256, 0, stream>>>(fdrug, NDRUG, dsim, drugSum, nullptr);
    }

    // shared = (gene@Wg + cna@Wc + mut@Wm + bg+bc+bm)/3 ; 512 wave-tiles -> 64 WGs
    linear_wmma3<<<(NCELL / 64) * (SHDIM / 16) / 8, 256, 0, stream>>>(
        geneN, Wg, GDIM, cnaN, Wc, CDIM, mutation, Wm, MDIM,
        bg, bc, bm, 1.f / 3.f, shrd, NCELL, SHDIM);
    // drug_embed = feature_drug @ Wd + bd ; 256 wave-tiles -> 32 WGs
    linear_wmma3<<<(NDRUG / 64) * (SHDIM / 16) / 8, 256, 0, stream>>>(
        fdrug, Wd, DDIM, nullptr, nullptr, 0, nullptr, nullptr, 0,
        bd, nullptr, nullptr, 1.f, dembed, NDRUG, SHDIM);

    dyn_mean<<<NCELL / 256, 256, 0, stream>>>(shrd, Wa1, ba1, Wa2, ba2, dynsum, NCELL);
    att_score<<<NCELL / 256, 256, 0, stream>>>(shrd,   NCELL, Wca1, bca1, Wca2, bca2, cellSc);
    att_score<<<NDRUG / 256, 256, 0, stream>>>(dembed, NDRUG, Wda1, bda1, Wda2, bda2, drugSc);

    finalize_coef<<<1, 32, 0, stream>>>(statw, dynsum, sumsq, coef, 1.f / (float)NCELL);

    topk_feature<<<NCELL, 256, 0, stream>>>(k0, k1, k2, coef, cellSc, WkC, bkC,
                                            out, NCELL, 0);
    topk_feature<<<NDRUG, 256, 0, stream>>>(dsim, nullptr, nullptr, coef, drugSc, WkD, bkD,
                                            out + (size_t)NCELL * SHDIM, NDRUG, 1);
}